// CAttention_49349174231386
// MI455X (gfx1250) — compile-verified
//
#include <hip/hip_runtime.h>
#include <math.h>

// ---------------------------------------------------------------------------
// CAttention for MI455X (gfx1250, wave32, WMMA).
// f32 WMMA (V_WMMA_F32_16X16X4_F32) for all GEMMs: exact f32 math matching
// the f32 reference.  attn@v_s is algebraically collapsed to a per-cluster
// reduction (exp(-1e9) underflows to 0 in f32, so each output row depends
// only on its cluster), turning 34 GFLOP into ~17 MFLOP.
// ---------------------------------------------------------------------------

#define BATCH 8
#define NTOK  2048
#define CDIM  512

typedef __attribute__((ext_vector_type(2))) float v2f;
typedef __attribute__((ext_vector_type(8))) float v8f;

// ---------------------------------------------------------------------------
// Generic f32 WMMA GEMM:  C = alpha * A@B (+ bias), optionally B transposed.
// Block = 256 threads (8 wave32s), tile 128(M) x 128(N) x 16(K).
// Wave grid 2x4, each wave computes 64x32 via 4x2 accumulators of 16x16.
// LDS: A tile [128][16] row-stride 20 (conflict-free b64 A-frag reads),
//      B tile [16][128] row-stride 136 (half-wave rows land in disjoint banks).
// ---------------------------------------------------------------------------
template <bool BT>
__global__ __launch_bounds__(256) void gemm_wmma_f32(
    const float* __restrict__ A, int lda,
    const float* __restrict__ Bm, int ldb,
    float* __restrict__ C, int ldc,
    int K, float alpha, const float* __restrict__ bias,
    long sA, long sB, long sC)
{
    __shared__ float As[128 * 20];
    __shared__ float Bs[16 * 136];

    const int tid  = threadIdx.x;
    const int lane = tid & 31;
    const int wave = tid >> 5;
    const int wm = (wave >> 2) * 64;   // wave M origin in tile
    const int wn = (wave & 3) * 32;    // wave N origin in tile
    const int m0 = blockIdx.y * 128;
    const int n0 = blockIdx.x * 128;

    const float* Ab = A  + (size_t)blockIdx.z * sA;
    const float* Bb = Bm + (size_t)blockIdx.z * sB;
    float*       Cb = C  + (size_t)blockIdx.z * sC;

    v8f acc[4][2] = {};

    const int rowA = tid >> 2;
    const int colA = (tid & 3) << 2;

    for (int k0 = 0; k0 < K; k0 += 16) {
        // ---- stage A tile: 128x16, two float4 per thread ----
        if (k0 + 16 < K)  // stream hint for the next stage
            __builtin_prefetch(&Ab[(size_t)(m0 + rowA) * lda + k0 + 16 + colA], 0, 0);
#pragma unroll
        for (int pp = 0; pp < 2; ++pp) {
            const int r = rowA + pp * 64;
            const float4 va = *(const float4*)&Ab[(size_t)(m0 + r) * lda + k0 + colA];
            *(float4*)&As[r * 20 + colA] = va;
        }
        // ---- stage B tile: 16x128 ----
        if (!BT) {
            const int kk = tid >> 4;
            const int nv = (tid & 15) << 2;
#pragma unroll
            for (int pp = 0; pp < 2; ++pp) {
                const int nn = nv + pp * 64;
                const float4 vb = *(const float4*)&Bb[(size_t)(k0 + kk) * ldb + n0 + nn];
                *(float4*)&Bs[kk * 136 + nn] = vb;
            }
        } else {
            // B given row-major over N (rows = output cols): transpose on store
            const int rB = tid >> 2;
            const int kv = (tid & 3) << 2;
#pragma unroll
            for (int pp = 0; pp < 2; ++pp) {
                const int nn = rB + pp * 64;
                const float4 vb = *(const float4*)&Bb[(size_t)(n0 + nn) * ldb + k0 + kv];
                Bs[(kv + 0) * 136 + nn] = vb.x;
                Bs[(kv + 1) * 136 + nn] = vb.y;
                Bs[(kv + 2) * 136 + nn] = vb.z;
                Bs[(kv + 3) * 136 + nn] = vb.w;
            }
        }
        __syncthreads();

        // ---- 4 WMMA K-steps of 4 over the 16-deep stage ----
#pragma unroll
        for (int kk = 0; kk < 16; kk += 4) {
            const int kb = kk + ((lane >> 4) << 1);  // A/B frag: K = 2*(lane/16)+vgpr
            v2f af[4];
#pragma unroll
            for (int mt = 0; mt < 4; ++mt) {
                const int m = wm + mt * 16 + (lane & 15);
                af[mt] = *(const v2f*)&As[m * 20 + kb];  // ds_load_b64, conflict-free
            }
            v2f bf[2];
#pragma unroll
            for (int nt = 0; nt < 2; ++nt) {
                const int n = wn + nt * 16 + (lane & 15);
                bf[nt].x = Bs[(kb + 0) * 136 + n];
                bf[nt].y = Bs[(kb + 1) * 136 + n];
            }
#pragma unroll
            for (int mt = 0; mt < 4; ++mt)
#pragma unroll
                for (int nt = 0; nt < 2; ++nt)
                    acc[mt][nt] = __builtin_amdgcn_wmma_f32_16x16x4_f32(
                        false, af[mt], false, bf[nt], (short)0, acc[mt][nt],
                        false, false);
        }
        __syncthreads();
    }

    // ---- epilogue: C/D layout -> VGPR r: lanes0-15 M=r, lanes16-31 M=r+8 ----
#pragma unroll
    for (int mt = 0; mt < 4; ++mt) {
        const int rbase = m0 + wm + mt * 16 + ((lane >> 4) << 3);
#pragma unroll
        for (int nt = 0; nt < 2; ++nt) {
            const int col = n0 + wn + nt * 16 + (lane & 15);
            const float bv = bias ? bias[col] : 0.0f;
#pragma unroll
            for (int r = 0; r < 8; ++r)
                Cb[(size_t)(rbase + r) * ldc + col] = acc[mt][nt][r] * alpha + bv;
        }
    }
}

// ---------------------------------------------------------------------------
// Stable counting argsort per batch over 16 cluster values.
// Produces idx_shuffle, idx_restore, sorted ids, and block boundaries.
// ---------------------------------------------------------------------------
__global__ __launch_bounds__(256) void sort_kernel(
    const int* __restrict__ idx, int* __restrict__ shuf, int* __restrict__ rest,
    int* __restrict__ ssort, int* __restrict__ cstart)
{
    __shared__ int chunk[16][256];
    __shared__ int off[17];
    __shared__ int total[16];
    const int b = blockIdx.x;
    const int t = threadIdx.x;
    const int* ic = idx + b * NTOK;

    int loc[8];
    int cnt[16];
#pragma unroll
    for (int c = 0; c < 16; ++c) cnt[c] = 0;
#pragma unroll
    for (int i = 0; i < 8; ++i) { loc[i] = ic[t * 8 + i] & 15; cnt[loc[i]]++; }
#pragma unroll
    for (int c = 0; c < 16; ++c) chunk[c][t] = cnt[c];
    __syncthreads();
    if (t < 16) {  // exclusive prefix over chunks for class t
        int base = 0;
        for (int u = 0; u < 256; ++u) { int tmp = chunk[t][u]; chunk[t][u] = base; base += tmp; }
        total[t] = base;
    }
    __syncthreads();
    if (t == 0) {
        int o = 0;
        for (int c = 0; c < 16; ++c) { off[c] = o; cstart[b * 17 + c] = o; o += total[c]; }
        off[16] = o; cstart[b * 17 + 16] = o;
    }
    __syncthreads();
    int offs[16];
#pragma unroll
    for (int c = 0; c < 16; ++c) offs[c] = off[c] + chunk[c][t];
#pragma unroll
    for (int i = 0; i < 8; ++i) {
        const int n = t * 8 + i;
        const int c = loc[i];
        const int p = offs[c]++;
        shuf[b * NTOK + p] = n;
        rest[b * NTOK + n] = p;
        ssort[b * NTOK + p] = c;
    }
}

// ---------------------------------------------------------------------------
// Per-batch cluster reduction:
//   e_j = exp(attn_map[b, shuf[shuf[j]], j])
//   out_c[ch] = (sum_{j in block c} e_j * v[shuf[j],ch] + (eps/N)*sum_j v[j,ch])
//               / (S_c + eps)
// Deterministic sequential per-thread-channel sums (no fp atomics).
// ---------------------------------------------------------------------------
__global__ __launch_bounds__(256) void cluster_reduce(
    const float* __restrict__ attn, const float* __restrict__ v,
    const int* __restrict__ shuf, const int* __restrict__ cstart,
    float* __restrict__ outc)
{
    __shared__ float e[NTOK];
    __shared__ float Ssum[16];
    __shared__ int cs[17];
    const int b = blockIdx.x;
    const int t = threadIdx.x;
    if (t < 17) cs[t] = cstart[b * 17 + t];
    const int* sh = shuf + b * NTOK;
    const float* am = attn + (size_t)b * NTOK * NTOK;
    for (int j = t; j < NTOK; j += 256) {
        const int js = sh[j];
        const int d  = sh[js];                 // double = shuf[shuf[j]]
        e[j] = expf(am[(size_t)d * NTOK + j]); // col_vals (already scaled)
    }
    __syncthreads();
    if (t < 16) {
        float s = 0.f;
        for (int j = cs[t]; j < cs[t + 1]; ++j) s += e[j];
        Ssum[t] = s;
    }
    __syncthreads();

    const float* vb = v + (size_t)b * NTOK * CDIM;
    const int ch0 = t, ch1 = t + 256;
    float vall0 = 0.f, vall1 = 0.f;
    for (int j = 0; j < NTOK; ++j) {
        vall0 += vb[(size_t)j * CDIM + ch0];
        vall1 += vb[(size_t)j * CDIM + ch1];
    }
    const float epsn = 1e-6f / (float)NTOK;
    for (int c = 0; c < 16; ++c) {
        float a0 = 0.f, a1 = 0.f;
        for (int j = cs[c]; j < cs[c + 1]; ++j) {
            const int row = sh[j];
            const float w = e[j];
            a0 += w * vb[(size_t)row * CDIM + ch0];
            a1 += w * vb[(size_t)row * CDIM + ch1];
        }
        const float den = Ssum[c] + 1e-6f;
        outc[((size_t)(b * 16 + c)) * CDIM + ch0] = (a0 + epsn * vall0) / den;
        outc[((size_t)(b * 16 + c)) * CDIM + ch1] = (a1 + epsn * vall1) / den;
    }
}

// ---------------------------------------------------------------------------
// out3[b,i,j] = out2[b, idx_restore[b,i], j] where out2 is the
// swapaxes(1,2).reshape permutation of the per-cluster-constant `out`:
//   out2[b,r,j] = outc[b, ssort[(r*512+j) mod 2048], (r*512+j) div 2048]
// ---------------------------------------------------------------------------
__global__ __launch_bounds__(256) void scramble(
    const float* __restrict__ outc, const int* __restrict__ rest,
    const int* __restrict__ ssort, float* __restrict__ out3)
{
    const int b = blockIdx.y;
    const int i = blockIdx.x;
    const int r = rest[b * NTOK + i];
    const int t = threadIdx.x;
#pragma unroll
    for (int pp = 0; pp < 2; ++pp) {
        const int j = t + pp * 256;
        const int p = r * CDIM + j;              // < 2048*512, fits int
        const int nsrc = p & (NTOK - 1);
        const int csrc = p >> 11;                // p / 2048
        out3[((size_t)(b * NTOK + i)) * CDIM + j] =
            outc[((size_t)(b * 16 + ssort[b * NTOK + nsrc])) * CDIM + csrc];
    }
}

// ---------------------------------------------------------------------------
extern "C" void kernel_launch(void* const* d_in, const int* in_sizes, int n_in,
                              void* d_out, int out_size, void* d_ws, size_t ws_size,
                              hipStream_t stream)
{
    (void)in_sizes; (void)n_in; (void)out_size; (void)ws_size;

    const float* x_token = (const float*)d_in[0];
    // d_in[1] x_path  : dead (pv never used)
    const int*   idx_cluster = (const int*)d_in[2];
    // d_in[3] cluster_num = 16; since ids are in [0,16) the (s<16) mask term
    // is always true — no device read needed.
    const float* Wqk   = (const float*)d_in[4];
    const float* Wv    = (const float*)d_in[5];
    // d_in[6] Wpv : dead
    const float* Wproj = (const float*)d_in[7];
    const float* bproj = (const float*)d_in[8];

    float* xout = (float*)d_out;                            // (8,2048,512)
    float* attn = xout + (size_t)BATCH * NTOK * CDIM;       // (8,2048,2048)

    // workspace layout (floats): qk 64MB | v 32MB | out3 32MB | outc | ints
    float* qk   = (float*)d_ws;
    float* v    = qk   + (size_t)BATCH * NTOK * 1024;
    float* out3 = v    + (size_t)BATCH * NTOK * CDIM;
    float* outc = out3 + (size_t)BATCH * NTOK * CDIM;
    int* shuf   = (int*)(outc + (size_t)BATCH * 16 * CDIM);
    int* rest   = shuf  + BATCH * NTOK;
    int* ssort  = rest  + BATCH * NTOK;
    int* cstart = ssort + BATCH * NTOK;

    const dim3 blk(256);
    const int M = BATCH * NTOK;  // 16384

    sort_kernel<<<dim3(BATCH), blk, 0, stream>>>(idx_cluster, shuf, rest, ssort, cstart);

    // qk = x_token @ Wqk   (16384 x 512 x 1024)
    gemm_wmma_f32<false><<<dim3(1024 / 128, M / 128, 1), blk, 0, stream>>>(
        x_token, 512, Wqk, 1024, qk, 1024, 512, 1.0f, nullptr, 0, 0, 0);

    // v = x_token @ Wv     (16384 x 512 x 512)
    gemm_wmma_f32<false><<<dim3(512 / 128, M / 128, 1), blk, 0, stream>>>(
        x_token, 512, Wv, 512, v, 512, 512, 1.0f, nullptr, 0, 0, 0);

    // attn_map[b] = SCALE * q[b] @ k[b]^T   (8 x 2048 x 2048 x 512, NT)
    gemm_wmma_f32<true><<<dim3(NTOK / 128, NTOK / 128, BATCH), blk, 0, stream>>>(
        qk, 1024, qk + 512, 1024, attn, NTOK, 512, 0.125f, nullptr,
        (long)NTOK * 1024, (long)NTOK * 1024, (long)NTOK * NTOK);

    // collapse attn @ v_s into per-cluster table
    cluster_reduce<<<dim3(BATCH), blk, 0, stream>>>(attn, v, shuf, cstart, outc);

    // permutation + restore gather -> out3
    scramble<<<dim3(NTOK, BATCH), blk, 0, stream>>>(outc, rest, ssort, out3);

    // x_out = out3 @ Wproj + bproj
    gemm_wmma_f32<false><<<dim3(512 / 128, M / 128, 1), blk, 0, stream>>>(
        out3, 512, Wproj, 512, xout, 512, 512, 1.0f, bproj, 0, 0, 0);
}